// Seq2Seq_42829413876381
// MI455X (gfx1250) — compile-verified
//
#include <hip/hip_runtime.h>
#include <math.h>

#define B      64
#define H      32
#define EMB_E  64
#define EMB_D  32
#define S_EN   50
#define S_DE   50
#define DE_V   19000
#define N_TILES 1188           // ceil(19000 / 16)
#define NBLK   64
#define NTHR   256
#define NWAVE  8               // wave32: 256/32

typedef __attribute__((ext_vector_type(2))) float v2f;
typedef __attribute__((ext_vector_type(8))) float v8f;

// ---------------- workspace layout (bytes) ----------------
// 0    : unsigned long long argA[2][B]   (1024)  double-buffered packed argmax
// 1024 : unsigned ctl[16]                (64)    ctl[0]=counter ctl[1]=generation
// 1088 : float hbuf[B*H]                 (8192)  decoder hidden state (global)
#define WS_CTL_OFF  1024
#define WS_H_OFF    1088
#define WS_INIT_WORDS ((WS_CTL_OFF + 64) / 4)

__device__ __forceinline__ float sigm(float x) { return 1.0f / (1.0f + expf(-x)); }

__device__ __forceinline__ unsigned long long shfl_xor_u64(unsigned long long x, int m) {
    unsigned lo = (unsigned)x;
    unsigned hi = (unsigned)(x >> 32);
    lo = (unsigned)__shfl_xor((int)lo, m, 32);
    hi = (unsigned)__shfl_xor((int)hi, m, 32);
    return ((unsigned long long)hi << 32) | lo;
}

// sense-reversing grid barrier (all NBLK blocks resident; wave32)
__device__ __forceinline__ void grid_barrier(unsigned* ctl) {
    __syncthreads();
    if (threadIdx.x == 0) {
        unsigned* cnt = ctl;
        unsigned* gen = ctl + 1;
        __threadfence();
        unsigned g = __hip_atomic_load(gen, __ATOMIC_RELAXED, __HIP_MEMORY_SCOPE_AGENT);
        unsigned prev = __hip_atomic_fetch_add(cnt, 1u, __ATOMIC_ACQ_REL, __HIP_MEMORY_SCOPE_AGENT);
        if (prev == NBLK - 1) {
            __hip_atomic_store(cnt, 0u, __ATOMIC_RELAXED, __HIP_MEMORY_SCOPE_AGENT);
            __hip_atomic_fetch_add(gen, 1u, __ATOMIC_RELEASE, __HIP_MEMORY_SCOPE_AGENT);
        } else {
            while (__hip_atomic_load(gen, __ATOMIC_ACQUIRE, __HIP_MEMORY_SCOPE_AGENT) == g)
                __builtin_amdgcn_s_sleep(2);
        }
        __threadfence();
    }
    __syncthreads();
}

__global__ void init_ws_kernel(unsigned* ws32, int nwords) {
    int i = blockIdx.x * blockDim.x + threadIdx.x;
    if (i < nwords) ws32[i] = 0u;
}

__global__ __launch_bounds__(NTHR, 1) void seq2seq_persistent(
    const int* __restrict__ en_batch, const int* __restrict__ en_lens,
    const int* __restrict__ de_batch,
    const float* __restrict__ en_emb,
    const float* __restrict__ eWih, const float* __restrict__ eWhh,
    const float* __restrict__ ebih, const float* __restrict__ ebhh,
    const float* __restrict__ de_emb,
    const float* __restrict__ dWih, const float* __restrict__ dWhh,
    const float* __restrict__ dbih, const float* __restrict__ dbhh,
    const float* __restrict__ fcW, const float* __restrict__ fcb,
    float* __restrict__ out,
    unsigned long long* __restrict__ argA, unsigned* __restrict__ ctl,
    float* __restrict__ hbuf)
{
    const int blk  = blockIdx.x;
    const int tid  = threadIdx.x;
    const int lane = tid & 31;
    const int w    = tid >> 5;          // wave index 0..7
    const int hi   = lane >> 4;         // lane-half (WMMA layout)
    const int ln   = lane & 15;

    __shared__ float s_h[NWAVE][H];     // per-wave LSTM hidden (blocks 0..7)
    __shared__ float s_c[NWAVE][H];
    __shared__ float s_hall[B][H + 1];  // all-rows h for WMMA A frags (+1 pad: bank spread)
    __shared__ unsigned long long s_arg[B];

    // ---- zero out[:, t=0, :] (grid-strided per row; NT stores: write-only stream) ----
    {
        const int t0 = blk * NTHR + tid;
        for (int b = 0; b < B; ++b) {
            float* orow0 = out + b * (S_DE * DE_V);
            for (int v = t0; v < DE_V; v += NBLK * NTHR)
                __builtin_nontemporal_store(0.0f, &orow0[v]);
        }
    }

    // ---- encoder: blocks 0..7, one wave per batch row ----
    if (blk < 8) {
        const int row = blk * 8 + w;
        const int j   = lane;           // H == 32 == wave32
        s_h[w][j] = 0.0f;
        s_c[w][j] = 0.0f;
        const int len = en_lens[row];
        for (int t = 0; t < S_EN; ++t) {
            const int tok = en_batch[row * S_EN + t];
            const float* x = en_emb + tok * EMB_E;
            float g[4];
            #pragma unroll
            for (int gi = 0; gi < 4; ++gi) {
                const int idx = gi * H + j;
                float acc = ebih[idx] + ebhh[idx];
                const float* wi = eWih + idx * EMB_E;
                #pragma unroll 8
                for (int k = 0; k < EMB_E; ++k) acc = fmaf(x[k], wi[k], acc);
                const float* wh = eWhh + idx * H;
                #pragma unroll 8
                for (int k = 0; k < H; ++k) acc = fmaf(s_h[w][k], wh[k], acc);
                g[gi] = acc;
            }
            const float i_ = sigm(g[0]), f_ = sigm(g[1]);
            const float gg = tanhf(g[2]), o_ = sigm(g[3]);
            const float cn = f_ * s_c[w][j] + i_ * gg;
            const float hn = o_ * tanhf(cn);
            if (t < len) { s_c[w][j] = cn; s_h[w][j] = hn; }
        }
        hbuf[row * H + j] = s_h[w][j];
    }
    grid_barrier(ctl);

    // ---- decoder: 49 greedy steps ----
    const int gw = blk * NWAVE + w;     // 0..511
    const int mt = gw & 3;              // fixed M-tile per wave

    for (int s = 1; s < S_DE; ++s) {
        // phase 1: LSTM step (blocks 0..7, one wave per row)
        if (blk < 8) {
            const int row = blk * 8 + w;
            const int j   = lane;
            int tok;
            if (s == 1) {
                tok = de_batch[row * S_DE];
            } else {
                unsigned long long p = argA[((s - 1) & 1) * B + row];
                tok = (int)(~(unsigned)(p & 0xFFFFFFFFull));
            }
            if (j == 0) argA[(s & 1) * B + row] = 0ull;   // reset buffer we are about to fill
            const float* x = de_emb + tok * EMB_D;
            float g[4];
            #pragma unroll
            for (int gi = 0; gi < 4; ++gi) {
                const int idx = gi * H + j;
                float acc = dbih[idx] + dbhh[idx];
                const float* wi = dWih + idx * EMB_D;
                #pragma unroll 8
                for (int k = 0; k < EMB_D; ++k) acc = fmaf(x[k], wi[k], acc);
                const float* wh = dWhh + idx * H;
                #pragma unroll 8
                for (int k = 0; k < H; ++k) acc = fmaf(s_h[w][k], wh[k], acc);
                g[gi] = acc;
            }
            const float i_ = sigm(g[0]), f_ = sigm(g[1]);
            const float gg = tanhf(g[2]), o_ = sigm(g[3]);
            const float cn = f_ * s_c[w][j] + i_ * gg;
            const float hn = o_ * tanhf(cn);
            s_c[w][j] = cn;
            s_h[w][j] = hn;
            hbuf[row * H + j] = hn;
        }
        grid_barrier(ctl);

        // phase 2: FC logits via fp32 WMMA + argmax (all blocks)
        for (int i = tid; i < B * H; i += NTHR) s_hall[i >> 5][i & 31] = hbuf[i];
        if (tid < B) s_arg[tid] = 0ull;
        __syncthreads();

        // A fragments (16x4 fp32 tile): VGPR0 = K {k0, k0+2}, VGPR1 = K {k0+1, k0+3}
        v2f a[8];
        {
            const int arow = mt * 16 + ln;
            #pragma unroll
            for (int kk = 0; kk < 8; ++kk) {
                const int kb = kk * 4 + 2 * hi;
                a[kk].x = s_hall[arow][kb];
                a[kk].y = s_hall[arow][kb + 1];
            }
        }

        unsigned long long best[8];
        #pragma unroll
        for (int r = 0; r < 8; ++r) best[r] = 0ull;

        for (int nt = gw >> 2; nt < N_TILES; nt += (NBLK * NWAVE) / 4) {
            const int v  = nt * 16 + ln;
            // Clamp instead of predicate: column N of D depends only on column N
            // of B, so out-of-range lanes may compute garbage — it is never
            // stored nor fed to the argmax (guarded below).
            const int vc = (v < DE_V) ? v : (DE_V - 1);
            const float* wrow = fcW + vc * H + 2 * hi;

            // 8 unconditional 8-byte B-fragment loads (clause, before the WMMAs)
            v2f bf[8];
            #pragma unroll
            for (int kk = 0; kk < 8; ++kk)
                bf[kk] = *(const v2f*)(wrow + kk * 4);
            const float bias = fcb[vc];

            // two independent accumulation chains to hide WMMA D->C latency
            v8f d0 = {0.f, 0.f, 0.f, 0.f, 0.f, 0.f, 0.f, 0.f};
            v8f d1 = {0.f, 0.f, 0.f, 0.f, 0.f, 0.f, 0.f, 0.f};
            #pragma unroll
            for (int kk = 0; kk < 4; ++kk) {
                d0 = __builtin_amdgcn_wmma_f32_16x16x4_f32(
                         false, a[2 * kk],     false, bf[2 * kk],     (short)0, d0, false, false);
                d1 = __builtin_amdgcn_wmma_f32_16x16x4_f32(
                         false, a[2 * kk + 1], false, bf[2 * kk + 1], (short)0, d1, false, false);
            }

            if (v < DE_V) {
                const int outcol = s * DE_V + v;
                #pragma unroll
                for (int r = 0; r < 8; ++r) {
                    const float val = d0[r] + d1[r] + bias;
                    const int orow = mt * 16 + r + 8 * hi;
                    // write-only 243MB stream -> non-temporal, keep L2 for fcW
                    __builtin_nontemporal_store(val, &out[orow * (S_DE * DE_V) + outcol]);
                    // monotonic float key; ~v in low bits so ties pick smallest index
                    unsigned ub  = __float_as_uint(val);
                    unsigned key = ub ^ (unsigned)(((int)ub >> 31) | 0x80000000);
                    unsigned long long p =
                        ((unsigned long long)key << 32) | (unsigned)(~(unsigned)v);
                    if (p > best[r]) best[r] = p;
                }
            }
        }

        // reduce argmax across the 16 lanes sharing each output row
        #pragma unroll
        for (int r = 0; r < 8; ++r) {
            unsigned long long b = best[r];
            #pragma unroll
            for (int m = 1; m < 16; m <<= 1) {
                unsigned long long o = shfl_xor_u64(b, m);
                if (o > b) b = o;
            }
            if (ln == 0) {
                const int orow = mt * 16 + r + 8 * hi;
                atomicMax(&s_arg[orow], b);
            }
        }
        __syncthreads();
        if (tid < B && s_arg[tid] != 0ull)
            atomicMax(&argA[(s & 1) * B + tid], s_arg[tid]);

        grid_barrier(ctl);
    }
}

extern "C" void kernel_launch(void* const* d_in, const int* in_sizes, int n_in,
                              void* d_out, int out_size, void* d_ws, size_t ws_size,
                              hipStream_t stream) {
    (void)in_sizes; (void)n_in; (void)out_size; (void)ws_size;
    const int*   en_batch = (const int*)  d_in[0];
    const int*   en_lens  = (const int*)  d_in[1];
    const int*   de_batch = (const int*)  d_in[2];
    const float* en_emb   = (const float*)d_in[3];
    const float* eWih     = (const float*)d_in[4];
    const float* eWhh     = (const float*)d_in[5];
    const float* ebih     = (const float*)d_in[6];
    const float* ebhh     = (const float*)d_in[7];
    const float* de_emb   = (const float*)d_in[8];
    const float* dWih     = (const float*)d_in[9];
    const float* dWhh     = (const float*)d_in[10];
    const float* dbih     = (const float*)d_in[11];
    const float* dbhh     = (const float*)d_in[12];
    const float* fcW      = (const float*)d_in[13];
    const float* fcb      = (const float*)d_in[14];

    float* out = (float*)d_out;
    unsigned long long* argA = (unsigned long long*)d_ws;
    unsigned* ctl  = (unsigned*)((char*)d_ws + WS_CTL_OFF);
    float*    hbuf = (float*)   ((char*)d_ws + WS_H_OFF);

    init_ws_kernel<<<1, 512, 0, stream>>>((unsigned*)d_ws, WS_INIT_WORDS);
    seq2seq_persistent<<<NBLK, NTHR, 0, stream>>>(
        en_batch, en_lens, de_batch, en_emb, eWih, eWhh, ebih, ebhh,
        de_emb, dWih, dWhh, dbih, dbhh, fcW, fcb,
        out, argA, ctl, hbuf);
}